// HCMGNNLayer_12300786335767
// MI455X (gfx1250) — compile-verified
//
#include <hip/hip_runtime.h>

#define D 128
#define LDSW 130   // padded LDS row stride (floats) -> conflict-free A-fragment reads

typedef __attribute__((ext_vector_type(2))) float v2f;
typedef __attribute__((ext_vector_type(8))) float v8f;

__device__ __forceinline__ v8f wmma_f32(v2f a, v2f b, v8f c) {
    // D = A(16x4,f32) x B(4x16,f32) + C(16x16,f32)
    return __builtin_amdgcn_wmma_f32_16x16x4_f32(false, a, false, b, (short)0, c,
                                                 false, false);
}

__global__ void zero_kernel(float* __restrict__ p, size_t n) {
    size_t i = (size_t)blockIdx.x * blockDim.x + threadIdx.x;
    if (i < n) p[i] = 0.0f;
}

// t[row][col] = sum_k x[row][k] * W[k][col] + bias[col]
// block = 256 threads = 8 waves; block handles 16 rows x 128 cols; wave w -> cols [16w,16w+16)
__global__ __launch_bounds__(256)
void lin_kernel(const float* __restrict__ x, const float* __restrict__ W,
                const float* __restrict__ bias, float* __restrict__ t) {
    __shared__ float aS[16 * LDSW];
    const int tid  = threadIdx.x;
    const int row0 = blockIdx.x * 16;

    for (int i = tid; i < 16 * D; i += 256) {
        int r = i >> 7, c = i & (D - 1);
        aS[r * LDSW + c] = x[(size_t)(row0 + r) * D + c];
    }
    __syncthreads();

    const int lane = tid & 31, wv = tid >> 5;
    const int n = lane & 15, h = lane >> 4, m = lane & 15;
    const int kb = h * 2;
    const int col = wv * 16 + n;

    v8f acc = {};
    for (int k = 0; k < D; k += 4) {
        v2f a = *(const v2f*)&aS[m * LDSW + k + kb];
        v2f b;
        b.x = W[(size_t)(k + kb) * D + col];
        b.y = W[(size_t)(k + kb + 1) * D + col];
        acc = wmma_f32(a, b, acc);
    }
    const float bv = bias[col];
    for (int p = 0; p < 8; ++p)
        t[(size_t)(row0 + p + 8 * h) * D + col] = acc[p] + bv;
}

// agg[dst] += tsrc[src] (per feature), cnt[dst] += 1 (per edge)
__global__ void scatter_kernel(const float* __restrict__ tsrc, const int* __restrict__ src,
                               const int* __restrict__ dst, float* __restrict__ agg,
                               float* __restrict__ cnt, int E) {
    size_t i = (size_t)blockIdx.x * blockDim.x + threadIdx.x;
    if (i >= (size_t)E * D) return;
    int e = (int)(i >> 7);
    int c = (int)(i & (D - 1));
    int s = src[e], d0 = dst[e];
    atomicAdd(&agg[(size_t)d0 * D + c], tsrc[(size_t)s * D + c]);
    if (c == 0) atomicAdd(&cnt[d0], 1.0f);
}

// Microbe dst: two SAGE branches (mm, hm) -> hetero mean -> residual -> LayerNorm
__global__ __launch_bounds__(256)
void combine_m_kernel(const float* __restrict__ t_m,
                      const float* __restrict__ agg_mm, const float* __restrict__ cnt_mm,
                      const float* __restrict__ agg_hm, const float* __restrict__ cnt_hm,
                      const float* __restrict__ Wl_mm, const float* __restrict__ bl_mm,
                      const float* __restrict__ Wr_mm,
                      const float* __restrict__ Wl_hm, const float* __restrict__ bl_hm,
                      const float* __restrict__ Wr_hm,
                      const float* __restrict__ g, const float* __restrict__ be,
                      float* __restrict__ out) {
    __shared__ float tS[16 * LDSW], aS[16 * LDSW], bS[16 * LDSW];
    __shared__ float red[4][16];
    const int tid  = threadIdx.x;
    const int row0 = blockIdx.x * 16;

    if (tid < 64) red[tid >> 4][tid & 15] = 0.0f;
    for (int i = tid; i < 16 * D; i += 256) {
        int r = i >> 7, c = i & (D - 1);
        int gr = row0 + r;
        tS[r * LDSW + c] = t_m[(size_t)gr * D + c];
        aS[r * LDSW + c] = agg_mm[(size_t)gr * D + c] / fmaxf(cnt_mm[gr], 1.0f);
        bS[r * LDSW + c] = agg_hm[(size_t)gr * D + c] / fmaxf(cnt_hm[gr], 1.0f);
    }
    __syncthreads();

    const int lane = tid & 31, wv = tid >> 5;
    const int n = lane & 15, h = lane >> 4, m = lane & 15;
    const int kb = h * 2;
    const int col = wv * 16 + n;

    v8f acc_mm = {}, acc_hm = {};
    for (int k = 0; k < D; k += 4) {
        v2f at = *(const v2f*)&tS[m * LDSW + k + kb];
        v2f aa = *(const v2f*)&aS[m * LDSW + k + kb];
        v2f ab = *(const v2f*)&bS[m * LDSW + k + kb];
        v2f b1; b1.x = Wl_mm[(size_t)(k + kb) * D + col]; b1.y = Wl_mm[(size_t)(k + kb + 1) * D + col];
        v2f b2; b2.x = Wr_mm[(size_t)(k + kb) * D + col]; b2.y = Wr_mm[(size_t)(k + kb + 1) * D + col];
        v2f b3; b3.x = Wl_hm[(size_t)(k + kb) * D + col]; b3.y = Wl_hm[(size_t)(k + kb + 1) * D + col];
        v2f b4; b4.x = Wr_hm[(size_t)(k + kb) * D + col]; b4.y = Wr_hm[(size_t)(k + kb + 1) * D + col];
        acc_mm = wmma_f32(aa, b1, acc_mm);
        acc_mm = wmma_f32(at, b2, acc_mm);
        acc_hm = wmma_f32(ab, b3, acc_hm);
        acc_hm = wmma_f32(at, b4, acc_hm);
    }

    float u1[8], u2[8];
    const float c1 = bl_mm[col], c2 = bl_hm[col];
    for (int p = 0; p < 8; ++p) {
        u1[p] = acc_mm[p] + c1;
        u2[p] = acc_hm[p] + c2;
        int r = p + 8 * h;
        atomicAdd(&red[0][r], u1[p] * u1[p]);
        atomicAdd(&red[1][r], u2[p] * u2[p]);
    }
    __syncthreads();

    float y[8];
    for (int p = 0; p < 8; ++p) {
        int r = p + 8 * h;
        float n1 = fmaxf(sqrtf(red[0][r]), 1e-12f);
        float n2 = fmaxf(sqrtf(red[1][r]), 1e-12f);
        float conv = 0.5f * (u1[p] / n1 + u2[p] / n2);
        y[p] = tS[r * LDSW + col] + conv;
        atomicAdd(&red[2][r], y[p]);
        atomicAdd(&red[3][r], y[p] * y[p]);
    }
    __syncthreads();

    const float gc = g[col], bc = be[col];
    for (int p = 0; p < 8; ++p) {
        int r = p + 8 * h;
        float mu  = red[2][r] * (1.0f / D);
        float var = red[3][r] * (1.0f / D) - mu * mu;
        out[(size_t)(row0 + r) * D + col] = (y[p] - mu) * rsqrtf(var + 1e-5f) * gc + bc;
    }
}

// Host dst: single SAGE branch (mh) -> residual -> LayerNorm
__global__ __launch_bounds__(256)
void combine_h_kernel(const float* __restrict__ t_h,
                      const float* __restrict__ agg_mh, const float* __restrict__ cnt_mh,
                      const float* __restrict__ Wl_mh, const float* __restrict__ bl_mh,
                      const float* __restrict__ Wr_mh,
                      const float* __restrict__ g, const float* __restrict__ be,
                      float* __restrict__ out) {
    __shared__ float tS[16 * LDSW], aS[16 * LDSW];
    __shared__ float red[3][16];
    const int tid  = threadIdx.x;
    const int row0 = blockIdx.x * 16;

    if (tid < 48) red[tid >> 4][tid & 15] = 0.0f;
    for (int i = tid; i < 16 * D; i += 256) {
        int r = i >> 7, c = i & (D - 1);
        int gr = row0 + r;
        tS[r * LDSW + c] = t_h[(size_t)gr * D + c];
        aS[r * LDSW + c] = agg_mh[(size_t)gr * D + c] / fmaxf(cnt_mh[gr], 1.0f);
    }
    __syncthreads();

    const int lane = tid & 31, wv = tid >> 5;
    const int n = lane & 15, h = lane >> 4, m = lane & 15;
    const int kb = h * 2;
    const int col = wv * 16 + n;

    v8f acc = {};
    for (int k = 0; k < D; k += 4) {
        v2f at = *(const v2f*)&tS[m * LDSW + k + kb];
        v2f aa = *(const v2f*)&aS[m * LDSW + k + kb];
        v2f b1; b1.x = Wl_mh[(size_t)(k + kb) * D + col]; b1.y = Wl_mh[(size_t)(k + kb + 1) * D + col];
        v2f b2; b2.x = Wr_mh[(size_t)(k + kb) * D + col]; b2.y = Wr_mh[(size_t)(k + kb + 1) * D + col];
        acc = wmma_f32(aa, b1, acc);
        acc = wmma_f32(at, b2, acc);
    }

    float u[8];
    const float c1 = bl_mh[col];
    for (int p = 0; p < 8; ++p) {
        u[p] = acc[p] + c1;
        atomicAdd(&red[0][p + 8 * h], u[p] * u[p]);
    }
    __syncthreads();

    float y[8];
    for (int p = 0; p < 8; ++p) {
        int r = p + 8 * h;
        float n1 = fmaxf(sqrtf(red[0][r]), 1e-12f);
        y[p] = tS[r * LDSW + col] + u[p] / n1;
        atomicAdd(&red[1][r], y[p]);
        atomicAdd(&red[2][r], y[p] * y[p]);
    }
    __syncthreads();

    const float gc = g[col], bc = be[col];
    for (int p = 0; p < 8; ++p) {
        int r = p + 8 * h;
        float mu  = red[1][r] * (1.0f / D);
        float var = red[2][r] * (1.0f / D) - mu * mu;
        out[(size_t)(row0 + r) * D + col] = (y[p] - mu) * rsqrtf(var + 1e-5f) * gc + bc;
    }
}

extern "C" void kernel_launch(void* const* d_in, const int* in_sizes, int n_in,
                              void* d_out, int out_size, void* d_ws, size_t ws_size,
                              hipStream_t stream) {
    const float* x_m    = (const float*)d_in[0];
    const float* x_h    = (const float*)d_in[1];
    const float* W_nt_m = (const float*)d_in[2];
    const float* b_nt_m = (const float*)d_in[3];
    const float* W_nt_h = (const float*)d_in[4];
    const float* b_nt_h = (const float*)d_in[5];
    const float* Wl_mm  = (const float*)d_in[6];
    const float* bl_mm  = (const float*)d_in[7];
    const float* Wr_mm  = (const float*)d_in[8];
    const float* Wl_mh  = (const float*)d_in[9];
    const float* bl_mh  = (const float*)d_in[10];
    const float* Wr_mh  = (const float*)d_in[11];
    const float* Wl_hm  = (const float*)d_in[12];
    const float* bl_hm  = (const float*)d_in[13];
    const float* Wr_hm  = (const float*)d_in[14];
    const float* g_m    = (const float*)d_in[15];
    const float* be_m   = (const float*)d_in[16];
    const float* g_h    = (const float*)d_in[17];
    const float* be_h   = (const float*)d_in[18];
    const int*   ei_mm  = (const int*)d_in[19];
    const int*   ei_mh  = (const int*)d_in[20];
    const int*   ei_hm  = (const int*)d_in[21];

    const int N_M  = in_sizes[0] / D;
    const int N_H  = in_sizes[1] / D;
    const int E_mm = in_sizes[19] / 2;
    const int E_mh = in_sizes[20] / 2;
    const int E_hm = in_sizes[21] / 2;

    const size_t nm = (size_t)N_M * D;
    const size_t nh = (size_t)N_H * D;

    float* ws     = (float*)d_ws;
    float* t_m    = ws;
    float* t_h    = t_m + nm;
    float* agg_mm = t_h + nh;
    float* agg_hm = agg_mm + nm;
    float* agg_mh = agg_hm + nm;
    float* cnt_mm = agg_mh + nh;
    float* cnt_hm = cnt_mm + N_M;
    float* cnt_mh = cnt_hm + N_M;

    // zero the contiguous accumulator region (aggs + counts) each call
    size_t zn = (nm + nm + nh) + (size_t)(N_M + N_M + N_H);
    zero_kernel<<<(int)((zn + 255) / 256), 256, 0, stream>>>(agg_mm, zn);

    // per-type linear transforms (WMMA f32 GEMM)
    lin_kernel<<<N_M / 16, 256, 0, stream>>>(x_m, W_nt_m, b_nt_m, t_m);
    lin_kernel<<<N_H / 16, 256, 0, stream>>>(x_h, W_nt_h, b_nt_h, t_h);

    // edge scatter (mean numerator + counts)
    auto sg = [](int E) { return (int)(((size_t)E * D + 255) / 256); };
    scatter_kernel<<<sg(E_mm), 256, 0, stream>>>(t_m, ei_mm, ei_mm + E_mm, agg_mm, cnt_mm, E_mm);
    scatter_kernel<<<sg(E_mh), 256, 0, stream>>>(t_m, ei_mh, ei_mh + E_mh, agg_mh, cnt_mh, E_mh);
    scatter_kernel<<<sg(E_hm), 256, 0, stream>>>(t_h, ei_hm, ei_hm + E_hm, agg_hm, cnt_hm, E_hm);

    // fused SAGE GEMMs + L2-normalize + hetero-mean + residual + LayerNorm
    float* out_m = (float*)d_out;
    float* out_h = out_m + nm;
    combine_m_kernel<<<N_M / 16, 256, 0, stream>>>(t_m, agg_mm, cnt_mm, agg_hm, cnt_hm,
                                                   Wl_mm, bl_mm, Wr_mm, Wl_hm, bl_hm, Wr_hm,
                                                   g_m, be_m, out_m);
    combine_h_kernel<<<N_H / 16, 256, 0, stream>>>(t_h, agg_mh, cnt_mh,
                                                   Wl_mh, bl_mh, Wr_mh,
                                                   g_h, be_h, out_h);
}